// AttentionComposition_39204461478341
// MI455X (gfx1250) — compile-verified
//
#include <hip/hip_runtime.h>
#include <hip/hip_bf16.h>
#include <stdint.h>

// MI455X / gfx1250, wave32.
// Sizes fixed by the reference: B=1024, L=128, W=H=512, NL=100.
#define B_SZ 1024
#define L_SZ 128
#define W_SZ 512

typedef __attribute__((ext_vector_type(2))) float        v2f;
typedef __attribute__((ext_vector_type(8))) float        v8f;
typedef __attribute__((ext_vector_type(4))) unsigned int v4u;
typedef __attribute__((ext_vector_type(8))) int          v8i;
typedef __attribute__((ext_vector_type(4))) int          v4i;

// ---------------------------------------------------------------------------
// Kernel 1: q = [nt_emb[nt_id] | stack_state] @ V_w   (B x 1024) x (1024 x 512)
// One wave computes one 16x16 tile of q via V_WMMA_F32_16X16X4_F32.
// A 16x4 f32 operand: lanes 0-15 hold M=0..15 {K, K+1}; lanes 16-31 hold {K+2, K+3}
//   -> per lane a contiguous float2 at (row, K + 2*hi).
// B 4x16 operand: same K split across lane halves, N = lane&15.
// ---------------------------------------------------------------------------
__global__ __launch_bounds__(128) void qproj_wmma(
    const float* __restrict__ nt_emb, const int* __restrict__ nt_id,
    const float* __restrict__ stack_state, const float* __restrict__ V_w,
    float* __restrict__ q_out)
{
    const int lane  = threadIdx.x & 31;
    const int wid   = blockIdx.x * 4 + (threadIdx.x >> 5);
    const int tileM = (wid >> 5) << 4;          // 64 tiles along B
    const int tileN = (wid & 31) << 4;          // 32 tiles along W
    const int hi    = lane >> 4;                // lane half selects K offset 0 / 2
    const int mrow  = tileM + (lane & 15);
    const int n     = tileN + (lane & 15);
    const int embBase = nt_id[mrow] * W_SZ;     // rhs cols [0,512): nt_emb gather
    const int stkBase = mrow * W_SZ;            // rhs cols [512,1024): stack_state
    v8f acc = {};
    for (int k = 0; k < 512; k += 4) {
        v2f a = *(const v2f*)(nt_emb + embBase + k + 2 * hi);
        v2f b;
        b.x = V_w[(k + 2 * hi) * W_SZ + n];
        b.y = V_w[(k + 2 * hi + 1) * W_SZ + n];
        acc = __builtin_amdgcn_wmma_f32_16x16x4_f32(false, a, false, b,
                                                    (short)0, acc, false, false);
    }
    for (int k = 0; k < 512; k += 4) {
        v2f a = *(const v2f*)(stack_state + stkBase + k + 2 * hi);
        v2f b;
        b.x = V_w[(512 + k + 2 * hi) * W_SZ + n];
        b.y = V_w[(512 + k + 2 * hi + 1) * W_SZ + n];
        acc = __builtin_amdgcn_wmma_f32_16x16x4_f32(false, a, false, b,
                                                    (short)0, acc, false, false);
    }
#pragma unroll
    for (int i = 0; i < 8; ++i)                 // C/D layout: VGPR i -> M = i + 8*hi
        q_out[(tileM + i + hi * 8) * W_SZ + n] = acc[i];
}

// ---------------------------------------------------------------------------
// Kernel 2: fused attention. One workgroup (8 waves) per batch row.
// children[b] (128x512 f32 = 256 KB) is staged into LDS with ONE Tensor Data
// Mover descriptor (2D tile, data_size=4B), then logits, softmax, and the
// weighted sum all run out of LDS -> children is read from HBM exactly once.
// q lives in the g-section of d_out and is overwritten in place (per-block)
// with weighted_child after it has been staged to LDS.
// ---------------------------------------------------------------------------
__global__ __launch_bounds__(256) void attn_fused(
    const float* __restrict__ children, const int* __restrict__ ch_lengths,
    float* q_wch,                        // in: q[b*512..], out: weighted_child
    float* __restrict__ attn_out)
{
    extern __shared__ float smem[];
    float* chs = smem;                   // 128 * 512 floats (256 KB)
    float* qs  = smem + L_SZ * W_SZ;     // 512 floats
    float* lgt = qs + W_SZ;              // 128 floats
    const int b    = blockIdx.x;
    const int tid  = threadIdx.x;
    const int lane = tid & 31;
    const int wv   = tid >> 5;

#if __has_builtin(__builtin_amdgcn_tensor_load_to_lds)
    if (wv == 0) {
        // D# group0: count=1 | lds_addr | global_addr | type=2
        unsigned lds_off = (unsigned)(uintptr_t)chs;   // low 32 bits = LDS offset
        unsigned long long ga =
            (unsigned long long)(uintptr_t)(children + (size_t)b * (L_SZ * W_SZ));
        v4u g0 = { 1u, lds_off, (unsigned)ga, (unsigned)(ga >> 32) | 0x80000000u };
        // D# group1: data_size=2 (4B); tensor_dim0=512, tensor_dim1=128,
        // tile_dim0=512, tile_dim1=128, tensor_dim0_stride=512.
        v8i g1 = { (int)(2u << 16), (int)(512u << 16), (int)(128u << 16),
                   (int)(512u << 16), 128, 512, 0, 0 };
        v4i gz = { 0, 0, 0, 0 };
#if __clang_major__ >= 23
        v8i gz8 = { 0, 0, 0, 0, 0, 0, 0, 0 };
        __builtin_amdgcn_tensor_load_to_lds(g0, g1, gz, gz, gz8, 0);
#else
        __builtin_amdgcn_tensor_load_to_lds(g0, g1, gz, gz, 0);
#endif
    }
    // stage q while the TDM streams children
    qs[tid]       = q_wch[b * W_SZ + tid];
    qs[tid + 256] = q_wch[b * W_SZ + tid + 256];
    if (wv == 0) __builtin_amdgcn_s_wait_tensorcnt(0);
#else
    qs[tid]       = q_wch[b * W_SZ + tid];
    qs[tid + 256] = q_wch[b * W_SZ + tid + 256];
    const float4* src = (const float4*)(children + (size_t)b * (L_SZ * W_SZ));
    float4* dst = (float4*)chs;
    for (int i = tid; i < (L_SZ * W_SZ) / 4; i += 256) dst[i] = src[i];
#endif
    __syncthreads();

    const int len = ch_lengths[b];
    // logits: wave wv handles rows wv, wv+8, ... ; lanes stride W -> conflict-free
    for (int l = wv; l < L_SZ; l += 8) {
        float p = 0.f;
#pragma unroll
        for (int i = 0; i < 16; ++i) {
            int w = lane + 32 * i;
            p += chs[l * W_SZ + w] * qs[w];
        }
#pragma unroll
        for (int off = 16; off > 0; off >>= 1) p += __shfl_xor(p, off, 32);
        if (lane == 0) lgt[l] = p;
    }
    __syncthreads();

    // masked softmax over 128 logits: single wave, 4 rows per lane
    if (wv == 0) {
        float m = -3.402823466e+38f;
#pragma unroll
        for (int j = 0; j < 4; ++j) {
            int l = lane + 32 * j;
            if (l < len) m = fmaxf(m, lgt[l]);
        }
#pragma unroll
        for (int off = 16; off > 0; off >>= 1) m = fmaxf(m, __shfl_xor(m, off, 32));
        float e[4], s = 0.f;
#pragma unroll
        for (int j = 0; j < 4; ++j) {
            int l = lane + 32 * j;
            e[j] = (l < len) ? __expf(lgt[l] - m) : 0.f;   // masked -> exp(-inf) = 0
            s += e[j];
        }
#pragma unroll
        for (int off = 16; off > 0; off >>= 1) s += __shfl_xor(s, off, 32);
        float rinv = 1.f / s;
#pragma unroll
        for (int j = 0; j < 4; ++j) lgt[lane + 32 * j] = e[j] * rinv;
    }
    __syncthreads();

    if (tid < L_SZ) attn_out[b * L_SZ + tid] = lgt[tid];

    // weighted_child: 2 columns per thread, broadcast attn weight from LDS
    float a0 = 0.f, a1 = 0.f;
    for (int l = 0; l < len; ++l) {
        float a = lgt[l];
        a0 += chs[l * W_SZ + tid] * a;
        a1 += chs[l * W_SZ + tid + 256] * a;
    }
    q_wch[b * W_SZ + tid]       = a0;   // overwrite consumed q in place
    q_wch[b * W_SZ + tid + 256] = a1;
}

// ---------------------------------------------------------------------------
// Kernel 3: gate GEMM. g_raw = sigmoid([nt_emb2[nt_id] | wch] @ gate_w.T + b)
// gate_w.T element (k,n) = gate_w[n*1024 + k] -> contiguous float2 B loads.
// Writes sigmoid output into the (free) c-section; combine happens in kernel 4
// so no block ever reads a wch row another block is overwriting.
// ---------------------------------------------------------------------------
__global__ __launch_bounds__(128) void gate_wmma(
    const float* __restrict__ nt_emb2, const int* __restrict__ nt_id,
    const float* __restrict__ wch, const float* __restrict__ gate_w,
    const float* __restrict__ gate_b, float* __restrict__ g_raw_out)
{
    const int lane  = threadIdx.x & 31;
    const int wid   = blockIdx.x * 4 + (threadIdx.x >> 5);
    const int tileM = (wid >> 5) << 4;
    const int tileN = (wid & 31) << 4;
    const int hi    = lane >> 4;
    const int mrow  = tileM + (lane & 15);
    const int n     = tileN + (lane & 15);
    const int embBase = nt_id[mrow] * W_SZ;
    const int wchBase = mrow * W_SZ;
    const int gwBase  = n * 1024;
    v8f acc = {};
    for (int k = 0; k < 512; k += 4) {
        v2f a = *(const v2f*)(nt_emb2 + embBase + k + 2 * hi);
        v2f b = *(const v2f*)(gate_w + gwBase + k + 2 * hi);
        acc = __builtin_amdgcn_wmma_f32_16x16x4_f32(false, a, false, b,
                                                    (short)0, acc, false, false);
    }
    for (int k = 0; k < 512; k += 4) {
        v2f a = *(const v2f*)(wch + wchBase + k + 2 * hi);
        v2f b = *(const v2f*)(gate_w + gwBase + 512 + k + 2 * hi);
        acc = __builtin_amdgcn_wmma_f32_16x16x4_f32(false, a, false, b,
                                                    (short)0, acc, false, false);
    }
    const float bias = gate_b[n];
#pragma unroll
    for (int i = 0; i < 8; ++i) {
        int row = tileM + i + hi * 8;
        float g = 1.f / (1.f + __expf(-(acc[i] + bias)));
        g_raw_out[row * W_SZ + n] = g;
    }
}

// ---------------------------------------------------------------------------
// Kernel 4: elementwise combine. c = g*nt2 + (1-g)*wch, then place g.
// Reads and writes each flat index exactly once per thread -> safe in-place.
// ---------------------------------------------------------------------------
__global__ __launch_bounds__(256) void compose_kernel(
    const float* __restrict__ nt_emb2, const int* __restrict__ nt_id,
    float* c_sec /* in: g_raw, out: c */, float* g_sec /* in: wch, out: g */)
{
    int idx = blockIdx.x * 256 + threadIdx.x;      // 0 .. B*W-1
    int row = idx >> 9;
    int col = idx & (W_SZ - 1);
    float g   = c_sec[idx];
    float w   = g_sec[idx];
    float nt2 = nt_emb2[nt_id[row] * W_SZ + col];
    c_sec[idx] = g * nt2 + (1.f - g) * w;
    g_sec[idx] = g;
}

// ---------------------------------------------------------------------------
// Launch. Inputs: children, ch_lengths, nt, nt_id, stack_state, V_w, nt_emb,
//                 nt_emb2, gate_w, gate_b.
// d_out layout: c[B*W] | attn[B*L] | g[B*W]  (all f32).
// ---------------------------------------------------------------------------
extern "C" void kernel_launch(void* const* d_in, const int* in_sizes, int n_in,
                              void* d_out, int out_size, void* d_ws, size_t ws_size,
                              hipStream_t stream) {
    const float* children    = (const float*)d_in[0];
    const int*   ch_lengths  = (const int*)d_in[1];
    const int*   nt_id       = (const int*)d_in[3];
    const float* stack_state = (const float*)d_in[4];
    const float* V_w         = (const float*)d_in[5];
    const float* nt_emb      = (const float*)d_in[6];
    const float* nt_emb2     = (const float*)d_in[7];
    const float* gate_w      = (const float*)d_in[8];
    const float* gate_b      = (const float*)d_in[9];

    float* out      = (float*)d_out;
    float* c_sec    = out;                           // B*W
    float* attn_sec = out + B_SZ * W_SZ;             // B*L
    float* g_sec    = out + B_SZ * W_SZ + B_SZ * L_SZ; // B*W (also q/wch staging)

    // 1) q -> g-section (64x32 tiles, 4 waves/block)
    qproj_wmma<<<512, 128, 0, stream>>>(nt_emb, nt_id, stack_state, V_w, g_sec);

    // 2) fused attention; 256 KB children tile + q + logits in LDS (264,704 B)
    size_t lds_bytes = (size_t)(L_SZ * W_SZ + W_SZ + L_SZ) * sizeof(float);
    attn_fused<<<B_SZ, 256, lds_bytes, stream>>>(children, ch_lengths, g_sec, attn_sec);

    // 3) gate GEMM -> sigmoid g into c-section
    gate_wmma<<<512, 128, 0, stream>>>(nt_emb2, nt_id, g_sec, gate_w, gate_b, c_sec);

    // 4) combine in place
    compose_kernel<<<(B_SZ * W_SZ) / 256, 256, 0, stream>>>(nt_emb2, nt_id, c_sec, g_sec);
}